// PyramidProposal_841813590618
// MI455X (gfx1250) — compile-verified
//
#include <hip/hip_runtime.h>
#include <cmath>

// ---------------- CDNA5 feature detection ----------------
#if defined(__has_builtin)
#  if __has_builtin(__builtin_amdgcn_global_load_async_to_lds_b128)
#    define ATH_ASYNC_LDS 1
#  endif
#  if __has_builtin(__builtin_amdgcn_s_wait_asynccnt)
#    define ATH_WAIT_ASYNC 1
#  endif
#  if __has_builtin(__builtin_amdgcn_wmma_f32_16x16x4_f32)
#    define ATH_WMMA 1
#  endif
#endif

// typed pointers for the async-to-LDS builtin (prototype per hipcc diagnostic)
typedef int v4i __attribute__((vector_size(16)));
typedef __attribute__((address_space(1))) v4i* gptr_v4i;
typedef __attribute__((address_space(3))) v4i* lptr_v4i;

// WMMA operand vector types (f32 16x16x4: A/B = 2 VGPRs, C/D = 8 VGPRs)
typedef __attribute__((ext_vector_type(2))) float v2f;
typedef __attribute__((ext_vector_type(8))) float v8f;

// ---------------- problem constants ----------------
#define BATCH   4
#define NANCH   255780            // total anchors per image
#define NPAD    262144            // padded stride (2^18)
#define PRE_NMS 6000
#define POST_NMS 1000
#define TOPP    6144              // padded top list (24*256)
#define COLW    192               // 6144/32 mask words per row
#define EQCAP   2192              // tie-bucket capacity (8192-6000)
#define NEGF    (-1000000000.0f)
#define NMS_TH  0.7f

struct Anchors { float v[5][3][4]; };

// order-preserving float<->u32 key
__device__ __forceinline__ unsigned f2k(float f) {
    unsigned b = __float_as_uint(f);
    return (b & 0x80000000u) ? ~b : (b | 0x80000000u);
}
__device__ __forceinline__ float k2f(unsigned k) {
    unsigned b = (k & 0x80000000u) ? (k ^ 0x80000000u) : ~k;
    return __uint_as_float(b);
}
// composite for ascending sort == (key desc, idx asc) descending order
__device__ __forceinline__ unsigned long long packAsc(unsigned key, unsigned idx) {
    return ~(((unsigned long long)key << 32) | (unsigned long long)(0xFFFFFFFFu - idx));
}

// ---------------- init ----------------
__global__ void init_kernel(unsigned long long* cand, unsigned* hist,
                            unsigned* prefix, unsigned* selK,
                            unsigned* cntGT, unsigned* cntEQ) {
    int i = blockIdx.x * 256 + threadIdx.x;
    if (i < BATCH * 8192) cand[i] = ~0ull;            // pads sort to the end
    if (i < 4 * BATCH * 256) hist[i] = 0u;
    if (i < BATCH) { prefix[i] = 0u; selK[i] = PRE_NMS; cntGT[i] = 0u; cntEQ[i] = 0u; }
}

// ---------------- decode ----------------
__global__ __launch_bounds__(256)
void decode_kernel(const float* __restrict__ c2, const float* __restrict__ b2,
                   const float* __restrict__ c3, const float* __restrict__ b3,
                   const float* __restrict__ c4, const float* __restrict__ b4,
                   const float* __restrict__ c5, const float* __restrict__ b5,
                   const float* __restrict__ c6, const float* __restrict__ b6,
                   const float* __restrict__ im_info, Anchors anc,
                   float* __restrict__ scores, float4* __restrict__ boxes) {
    int n = blockIdx.x * 256 + threadIdx.x;
    int b = blockIdx.y;
    if (n >= NANCH) return;

    int l, rel, Hh, Ww, st; const float* cp; const float* bp;
    if      (n < 192000) { l=0; rel=n;        Hh=200; Ww=320; st=4;  cp=c2; bp=b2; }
    else if (n < 240000) { l=1; rel=n-192000; Hh=100; Ww=160; st=8;  cp=c3; bp=b3; }
    else if (n < 252000) { l=2; rel=n-240000; Hh=50;  Ww=80;  st=16; cp=c4; bp=b4; }
    else if (n < 255000) { l=3; rel=n-252000; Hh=25;  Ww=40;  st=32; cp=c5; bp=b5; }
    else                 { l=4; rel=n-255000; Hh=13;  Ww=20;  st=64; cp=c6; bp=b6; }
    int p = rel / 3, a = rel % 3;
    int h = p / Ww, w = p % Ww;

    float sc = cp[(((size_t)b * 6 + 3 + a) * Hh + h) * Ww + w];
    size_t plane = (size_t)Hh * Ww;
    size_t bb0 = (((size_t)b * 12 + 4 * a) * Hh + h) * Ww + w;
    float d0 = bp[bb0], d1 = bp[bb0 + plane], d2 = bp[bb0 + 2*plane], d3 = bp[bb0 + 3*plane];

    float sx = (float)(st * w), sy = (float)(st * h);
    float ax1 = anc.v[l][a][0] + sx, ay1 = anc.v[l][a][1] + sy;
    float ax2 = anc.v[l][a][2] + sx, ay2 = anc.v[l][a][3] + sy;
    float aw = ax2 - ax1 + 1.0f, ah = ay2 - ay1 + 1.0f;
    float acx = ax1 + 0.5f * (aw - 1.0f), acy = ay1 + 0.5f * (ah - 1.0f);
    float pcx = d0 * aw + acx, pcy = d1 * ah + acy;
    float pw = __expf(d2) * aw, ph = __expf(d3) * ah;
    float x1 = pcx - 0.5f * (pw - 1.0f), y1 = pcy - 0.5f * (ph - 1.0f);
    float x2 = pcx + 0.5f * (pw - 1.0f), y2 = pcy + 0.5f * (ph - 1.0f);

    float wlim = im_info[b*3+1] - 1.0f, hlim = im_info[b*3+0] - 1.0f;
    x1 = fminf(fmaxf(x1, 0.0f), wlim); x2 = fminf(fmaxf(x2, 0.0f), wlim);
    y1 = fminf(fmaxf(y1, 0.0f), hlim); y2 = fminf(fmaxf(y2, 0.0f), hlim);
    float ms = 0.0f * im_info[b*3+2];   // MIN_SIZE = 0
    bool valid = (x2 - x1 + 1.0f >= ms) && (y2 - y1 + 1.0f >= ms);

    scores[(size_t)b * NPAD + n] = valid ? sc : NEGF;
    boxes [(size_t)b * NPAD + n] = make_float4(x1, y1, x2, y2);
}

// ---------------- radix-select top-6000 ----------------
__global__ __launch_bounds__(256)
void hist_kernel(const float* __restrict__ scores, const unsigned* __restrict__ prefix,
                 unsigned* __restrict__ hist, int pass) {
    int b = blockIdx.y, tid = threadIdx.x;
    __shared__ unsigned h[256];
    h[tid] = 0u; __syncthreads();
    unsigned pref = prefix[b];
    int base = blockIdx.x * 1024;
    for (int k = 0; k < 4; ++k) {
        int n = base + k * 256 + tid;
        if (n < NANCH) {
            unsigned key = f2k(scores[(size_t)b * NPAD + n]);
            bool ok = (pass == 0) || ((key >> (32 - 8 * pass)) == pref);
            if (ok) atomicAdd(&h[(key >> (24 - 8 * pass)) & 255u], 1u);
        }
    }
    __syncthreads();
    if (h[tid]) atomicAdd(&hist[((size_t)pass * BATCH + b) * 256 + tid], h[tid]);
}

__global__ void select_kernel(unsigned* prefix, unsigned* selK,
                              const unsigned* __restrict__ hist, int pass) {
    int b = threadIdx.x;
    if (b >= BATCH) return;
    unsigned K = selK[b], pref = prefix[b];
    const unsigned* h = hist + ((size_t)pass * BATCH + b) * 256;
    int v = 255;
    for (; v >= 0; --v) { unsigned c = h[v]; if (K <= c) break; K -= c; }
    if (v < 0) v = 0;
    prefix[b] = (pref << 8) | (unsigned)v;
    selK[b] = K;   // after pass 3: #ties at exact key to take (lowest index first)
}

__global__ __launch_bounds__(256)
void compact_kernel(const float* __restrict__ scores, const unsigned* __restrict__ thresh,
                    unsigned* cntGT, unsigned* cntEQ, unsigned long long* __restrict__ cand) {
    int n = blockIdx.x * 256 + threadIdx.x;
    int b = blockIdx.y;
    if (n >= NANCH) return;
    unsigned key = f2k(scores[(size_t)b * NPAD + n]);
    unsigned T = thresh[b];
    if (key > T) {
        unsigned pos = atomicAdd(&cntGT[b], 1u);            // total < 6000 guaranteed
        cand[(size_t)b * 8192 + pos] = packAsc(key, (unsigned)n);
    } else if (key == T) {
        unsigned pos = atomicAdd(&cntEQ[b], 1u);
        if (pos < EQCAP) cand[(size_t)b * 8192 + PRE_NMS + pos] = packAsc(key, (unsigned)n);
    }
}

// ---------------- bitonic sort (ascending on composite) ----------------
template<int NTOT, int NTHR>
__device__ inline void bitonic_sort_asc(unsigned long long* sm) {
    int tid = threadIdx.x;
    for (int k = 2; k <= NTOT; k <<= 1) {
        for (int j = k >> 1; j > 0; j >>= 1) {
            __syncthreads();
            for (int t = tid; t < NTOT; t += NTHR) {
                int ixj = t ^ j;
                if (ixj > t) {
                    unsigned long long a = sm[t], c = sm[ixj];
                    bool up = ((t & k) == 0);
                    if (up ? (a > c) : (a < c)) { sm[t] = c; sm[ixj] = a; }
                }
            }
        }
    }
    __syncthreads();
}

__global__ __launch_bounds__(1024)
void sort_topk_kernel(const unsigned long long* __restrict__ cand,
                      const float4* __restrict__ boxes,
                      float* __restrict__ topS, float4* __restrict__ topB) {
    int b = blockIdx.x, tid = threadIdx.x;
    __shared__ unsigned long long sm[8192];   // 64 KB of the 320 KB WGP LDS
    for (int i = tid; i < 8192; i += 1024) sm[i] = cand[(size_t)b * 8192 + i];
    __syncthreads();
    bitonic_sort_asc<8192, 1024>(sm);
    for (int i = tid; i < TOPP; i += 1024) {
        float sc = NEGF;
        float4 bx = make_float4(-1e8f, -1e8f, -1e8f, -1e8f);   // far-away sentinel
        if (i < PRE_NMS) {
            unsigned long long c = ~sm[i];
            unsigned key = (unsigned)(c >> 32);
            unsigned idx = 0xFFFFFFFFu - (unsigned)c;
            if (idx < (unsigned)NANCH) { sc = k2f(key); bx = boxes[(size_t)b * NPAD + idx]; }
        }
        topS[(size_t)b * TOPP + i] = sc;
        topB[(size_t)b * TOPP + i] = bx;
    }
}

// ---------------- NMS pairwise bitmask ----------------
// Wave-centric: each wave32 owns 16 rows. Per 32-col subtile, two
// V_WMMA_F32_16X16X4_F32 compute the 16x16 denominator tiles
// S[i][j] = area_i + area_j  (rank-2 matmul: [a,1] x [1^T; a^T]),
// co-executing with the VALU min/max intersection work. Column boxes are
// staged into LDS with GLOBAL_LOAD_ASYNC_TO_LDS_B128.
__global__ __launch_bounds__(256)
void nms_mask_kernel(const float4* __restrict__ topB, unsigned* __restrict__ mask) {
    int b = blockIdx.z;
    int rowBase = blockIdx.x * 128;           // 47 * 128 == 6016 rows exactly
    int colBase = blockIdx.y * 256;           // 24 * 256 == 6144 cols exactly
    int tid = threadIdx.x, wave = tid >> 5, lane = tid & 31;
    __shared__ float4 colTile[256];
    __shared__ float4 rowTile[128];

    const float4* csrc = topB + (size_t)b * TOPP + colBase + tid;
#if defined(ATH_ASYNC_LDS)
    __builtin_amdgcn_global_load_async_to_lds_b128(
        (gptr_v4i)(csrc), (lptr_v4i)(&colTile[tid]), 0, 0);
#else
    colTile[tid] = *csrc;
#endif
    if (tid < 128) rowTile[tid] = topB[(size_t)b * TOPP + rowBase + tid];
#if defined(ATH_ASYNC_LDS)
#  if defined(ATH_WAIT_ASYNC)
    __builtin_amdgcn_s_wait_asynccnt(0);
#  else
    asm volatile("s_wait_asynccnt 0" ::: "memory");
#  endif
#endif
    __syncthreads();

    int rb = wave * 16;                        // wave's rows within rowTile
    int laneHi = lane >> 4, lane15 = lane & 15;

    // A operand: lanes 0-15 hold A[M=lane][K0]=area_row, [K1]=1 ; lanes 16-31 K2,K3 = 0
    float rArea0 = 0.0f;
    {
        float4 rx = rowTile[rb + lane15];
        rArea0 = (rx.z - rx.x + 1.0f) * (rx.w - rx.y + 1.0f);
    }
    v2f Aop;
    Aop.x = (lane < 16) ? rArea0 : 0.0f;
    Aop.y = (lane < 16) ? 1.0f   : 0.0f;

    // the 8 row boxes this lane compares against (rows r + 8*laneHi)
    float4 rbox[8];
    #pragma unroll
    for (int r = 0; r < 8; ++r) rbox[r] = rowTile[rb + r + 8 * laneHi];

    for (int s = 0; s < 8; ++s) {              // 8 subtiles of 32 columns
        unsigned bal[2][8];
        #pragma unroll
        for (int hh = 0; hh < 2; ++hh) {       // two 16-wide halves
            int cloc = s * 32 + hh * 16;
            float4 cx = colTile[cloc + lane15];
            float cArea = (cx.z - cx.x + 1.0f) * (cx.w - cx.y + 1.0f);
#if defined(ATH_WMMA)
            // B operand: v0 lanes 0-15 = row K0 (ones), lanes 16-31 = row K1 (col areas); v1 = 0
            v2f Bop;
            Bop.x = (lane < 16) ? 1.0f : cArea;
            Bop.y = 0.0f;
            v8f Cz = {};
            v8f D = __builtin_amdgcn_wmma_f32_16x16x4_f32(
                false, Aop, false, Bop, (short)0, Cz, false, false);
#endif
            #pragma unroll
            for (int r = 0; r < 8; ++r) {
                float4 rx = rbox[r];
                float xx1 = fmaxf(rx.x, cx.x), yy1 = fmaxf(rx.y, cx.y);
                float xx2 = fminf(rx.z, cx.z), yy2 = fminf(rx.w, cx.w);
                float iw = fmaxf(xx2 - xx1 + 1.0f, 0.0f);
                float ih = fmaxf(yy2 - yy1 + 1.0f, 0.0f);
                float inter = iw * ih;
#if defined(ATH_WMMA)
                float S = D[r];                // area_row + area_col from the matrix pipe
#else
                float rA = (rx.z - rx.x + 1.0f) * (rx.w - rx.y + 1.0f);
                float S = rA + cArea;
#endif
                float iou = inter / (S - inter);
                bal[hh][r] = (unsigned)__ballot(iou > NMS_TH);
            }
        }
        // assemble one 32-col word per row; lanes 0-15 write rows rowBase+rb+lane
        int rowG = rowBase + rb + lane15;
        int c0 = colBase + s * 32;
        int rr = lane15 & 7;
        unsigned w0;
        if (lane15 < 8) w0 = (bal[0][rr] & 0xFFFFu) | (bal[1][rr] << 16);
        else            w0 = (bal[0][rr] >> 16)     | (bal[1][rr] & 0xFFFF0000u);
        int d = rowG - c0 + 1;                 // only cols strictly greater than row
        unsigned allowed = (d <= 0) ? 0xFFFFFFFFu : ((d >= 32) ? 0u : ~((1u << d) - 1u));
        if (lane < 16 && rowG < PRE_NMS) {
            mask[((size_t)b * PRE_NMS + rowG) * COLW + (colBase >> 5) + s] = w0 & allowed;
        }
    }
}

// ---------------- serial greedy collect: one wave32 per image ----------------
__global__ __launch_bounds__(32)
void collect_kernel(const unsigned* __restrict__ mask, unsigned* __restrict__ supp) {
    int b = blockIdx.x, lane = threadIdx.x;
    __shared__ unsigned rem[COLW];
    for (int w = lane; w < COLW; w += 32) rem[w] = 0u;
    __syncthreads();
    volatile unsigned* vrem = rem;
    for (int i = 0; i < PRE_NMS; ++i) {
        __builtin_prefetch(mask + ((size_t)b * PRE_NMS + (i + 1)) * COLW + lane * 6, 0, 0);
        unsigned rword = vrem[i >> 5];
        unsigned s = (rword >> (i & 31)) & 1u;
        if (lane == 0) supp[(size_t)b * 6016 + i] = s;
        if (!s) {
            const unsigned* row = mask + ((size_t)b * PRE_NMS + i) * COLW;
            for (int w = lane; w < COLW; w += 32) vrem[w] |= row[w];
        }
    }
}

// ---------------- post-NMS top-1000 via wave32 ballot scan ----------------
__global__ __launch_bounds__(32)
void postsel_kernel(const unsigned* __restrict__ supp,
                    const float* __restrict__ topS, const float4* __restrict__ topB,
                    float* __restrict__ postS, float4* __restrict__ postB) {
    int b = blockIdx.x, lane = threadIdx.x;
    unsigned lt = (1u << lane) - 1u;
    int NS = 0;
    for (int base = 0; base < PRE_NMS; base += 32) {
        int i = base + lane;
        bool keep = (i < PRE_NMS) && (supp[(size_t)b * 6016 + i] == 0u);
        NS += __popc((unsigned)__ballot(keep));
    }
    int nsPos = 0, sPos = NS;
    for (int base = 0; base < PRE_NMS; base += 32) {
        int i = base + lane;
        bool valid = (i < PRE_NMS);
        unsigned sp = valid ? supp[(size_t)b * 6016 + i] : 0u;
        unsigned blk = (unsigned)__ballot(valid && !sp);
        unsigned bls = (unsigned)__ballot(valid && sp);
        if (valid) {
            int pos = sp ? (sPos + __popc(bls & lt)) : (nsPos + __popc(blk & lt));
            if (pos < POST_NMS) {
                postS[(size_t)b * POST_NMS + pos] = sp ? NEGF : topS[(size_t)b * TOPP + i];
                postB[(size_t)b * POST_NMS + pos] = topB[(size_t)b * TOPP + i];
            }
        }
        nsPos += __popc(blk);
        sPos  += __popc(bls);
    }
}

// ---------------- final global stable sort + ROI emit ----------------
__global__ __launch_bounds__(1024)
void final_kernel(const float* __restrict__ postS, const float4* __restrict__ postB,
                  float* __restrict__ out) {
    int tid = threadIdx.x;
    __shared__ unsigned long long sm[4096];   // 32 KB LDS
    for (int i = tid; i < 4096; i += 1024) {
        unsigned long long v = ~0ull;
        if (i < BATCH * POST_NMS) v = packAsc(f2k(postS[i]), (unsigned)i);
        sm[i] = v;
    }
    __syncthreads();
    bitonic_sort_asc<4096, 1024>(sm);
    for (int j = tid; j < POST_NMS; j += 1024) {
        unsigned long long c = ~sm[j];
        unsigned key = (unsigned)(c >> 32);
        unsigned g = 0xFFFFFFFFu - (unsigned)c;
        float4 bx = postB[g];
        int bimg = (int)(g / POST_NMS);
        out[j * 5 + 0] = (float)bimg;
        out[j * 5 + 1] = bx.x; out[j * 5 + 2] = bx.y;
        out[j * 5 + 3] = bx.z; out[j * 5 + 4] = bx.w;
        out[5 * POST_NMS + j] = k2f(key);
    }
}

// ---------------- host side ----------------
static void make_anchors(Anchors& A) {
    const int strides[5] = {4, 8, 16, 32, 64};
    const double ratios[3] = {0.5, 1.0, 2.0};
    for (int l = 0; l < 5; ++l) {
        double bs = (double)strides[l];
        double size = bs * bs;
        double c = 0.5 * (bs - 1.0);
        for (int r = 0; r < 3; ++r) {
            double ws = std::nearbyint(std::sqrt(size / ratios[r]));   // half-even == np.round
            double hs = std::nearbyint(ws * ratios[r]);
            double w2 = ws * 8.0, h2 = hs * 8.0;
            A.v[l][r][0] = (float)(c - 0.5 * (w2 - 1.0));
            A.v[l][r][1] = (float)(c - 0.5 * (h2 - 1.0));
            A.v[l][r][2] = (float)(c + 0.5 * (w2 - 1.0));
            A.v[l][r][3] = (float)(c + 0.5 * (h2 - 1.0));
        }
    }
}

extern "C" void kernel_launch(void* const* d_in, const int* in_sizes, int n_in,
                              void* d_out, int out_size, void* d_ws, size_t ws_size,
                              hipStream_t stream) {
    (void)in_sizes; (void)n_in; (void)out_size; (void)ws_size;
    // setup_inputs dict order: cls_p2,bbox_p2, cls_p3,bbox_p3, ..., cls_p6,bbox_p6, im_info
    const float* cls[5]; const float* bbx[5];
    for (int l = 0; l < 5; ++l) { cls[l] = (const float*)d_in[2*l]; bbx[l] = (const float*)d_in[2*l+1]; }
    const float* im_info = (const float*)d_in[10];

    Anchors anc; make_anchors(anc);

    char* ws = (char*)d_ws; size_t off = 0;
    auto carve = [&](size_t bytes) -> char* {
        char* p = ws + off;
        off = (off + bytes + 255) & ~(size_t)255;
        return p;
    };
    float*              scores = (float*)             carve((size_t)BATCH * NPAD * 4);
    float4*             boxes  = (float4*)            carve((size_t)BATCH * NPAD * 16);
    unsigned*           hist   = (unsigned*)          carve((size_t)4 * BATCH * 256 * 4);
    unsigned*           prefix = (unsigned*)          carve(256);
    unsigned*           selK   = (unsigned*)          carve(256);
    unsigned*           cntGT  = (unsigned*)          carve(256);
    unsigned*           cntEQ  = (unsigned*)          carve(256);
    unsigned long long* cand   = (unsigned long long*)carve((size_t)BATCH * 8192 * 8);
    float*              topS   = (float*)             carve((size_t)BATCH * TOPP * 4);
    float4*             topB   = (float4*)            carve((size_t)BATCH * TOPP * 16);
    unsigned*           mask   = (unsigned*)          carve((size_t)BATCH * PRE_NMS * COLW * 4);
    unsigned*           supp   = (unsigned*)          carve((size_t)BATCH * 6016 * 4);
    float*              postS  = (float*)             carve((size_t)BATCH * POST_NMS * 4);
    float4*             postB  = (float4*)            carve((size_t)BATCH * POST_NMS * 16);

    init_kernel<<<128, 256, 0, stream>>>(cand, hist, prefix, selK, cntGT, cntEQ);

    decode_kernel<<<dim3(1000, BATCH), 256, 0, stream>>>(
        cls[0], bbx[0], cls[1], bbx[1], cls[2], bbx[2], cls[3], bbx[3], cls[4], bbx[4],
        im_info, anc, scores, boxes);

    for (int p = 0; p < 4; ++p) {
        hist_kernel<<<dim3(250, BATCH), 256, 0, stream>>>(scores, prefix, hist, p);
        select_kernel<<<1, 32, 0, stream>>>(prefix, selK, hist, p);
    }
    compact_kernel<<<dim3(1000, BATCH), 256, 0, stream>>>(scores, prefix, cntGT, cntEQ, cand);

    sort_topk_kernel<<<BATCH, 1024, 0, stream>>>(cand, boxes, topS, topB);

    nms_mask_kernel<<<dim3(47, 24, BATCH), 256, 0, stream>>>(topB, mask);
    collect_kernel<<<BATCH, 32, 0, stream>>>(mask, supp);
    postsel_kernel<<<BATCH, 32, 0, stream>>>(supp, topS, topB, postS, postB);

    final_kernel<<<1, 1024, 0, stream>>>(postS, postB, (float*)d_out);
}